// QNN_QSquared_54898271977539
// MI455X (gfx1250) — compile-verified
//
#include <hip/hip_runtime.h>
#include <math.h>

// ---- fixed problem shape ----
#define NQ 8
#define DIMQ 256      // 2^NQ state dim
#define NBATCH 256
#define TSTEPS 8
#define NBLK 8
#define NLAYERS 2
#define QDEPTH 2
#define NCIRC 64      // TSTEPS*NBLK circuits per layer
#define KBLOCKS 16
#define MT 16         // U-rows per workgroup (k-block size)

typedef float v2f __attribute__((ext_vector_type(2)));
typedef float v8f __attribute__((ext_vector_type(8)));

struct c32 { float x, y; };
__device__ __forceinline__ c32 cmul(c32 a, c32 b) { return {a.x*b.x - a.y*b.y, a.x*b.y + a.y*b.x}; }
__device__ __forceinline__ c32 cadd(c32 a, c32 b) { return {a.x + b.x, a.y + b.y}; }
__device__ __forceinline__ c32 shfl_xor_c(c32 a, int m) {
    c32 r; r.x = __shfl_xor(a.x, m, 32); r.y = __shfl_xor(a.y, m, 32); return r;
}

// ---------------- gate application on a register-resident 256-dim state ----------------
// Layout: lane L of the wave holds amplitudes k = s*32 + L for s=0..7 (wire w <-> bit 7-w of k).

__device__ __forceinline__ void rot_g(c32* st, int bit, const float* g, int lane)
{
    c32 g00 = {g[0], g[1]}, g01 = {g[2], g[3]}, g10 = {g[4], g[5]}, g11 = {g[6], g[7]};
    if (bit >= 5) {
        const int m = 1 << (bit - 5);
        #pragma unroll
        for (int s = 0; s < 8; ++s) {
            if (!(s & m)) {
                c32 a = st[s], b = st[s | m];
                st[s]     = cadd(cmul(g00, a), cmul(g01, b));
                st[s | m] = cadd(cmul(g10, a), cmul(g11, b));
            }
        }
    } else {
        const int mask = 1 << bit;
        const bool hi = (lane >> bit) & 1;
        #pragma unroll
        for (int s = 0; s < 8; ++s) {
            c32 mine = st[s];
            c32 o = shfl_xor_c(mine, mask);
            c32 lo_v = cadd(cmul(g00, mine), cmul(g01, o));
            c32 hi_v = cadd(cmul(g10, o), cmul(g11, mine));
            st[s].x = hi ? hi_v.x : lo_v.x;
            st[s].y = hi ? hi_v.y : lo_v.y;
        }
    }
}

__device__ __forceinline__ void cnot_g(c32* st, int bc, int bt, int lane)
{
    if (bt >= 5) {
        const int tm = 1 << (bt - 5);
        if (bc >= 5) {
            const int cm = 1 << (bc - 5);
            #pragma unroll
            for (int s = 0; s < 8; ++s) {
                if (!(s & tm) && (s & cm)) { c32 t = st[s]; st[s] = st[s | tm]; st[s | tm] = t; }
            }
        } else {
            const bool ctrl = (lane >> bc) & 1;
            #pragma unroll
            for (int s = 0; s < 8; ++s) {
                if (!(s & tm)) {
                    c32 a = st[s], b = st[s | tm];
                    st[s].x = ctrl ? b.x : a.x; st[s].y = ctrl ? b.y : a.y;
                    st[s | tm].x = ctrl ? a.x : b.x; st[s | tm].y = ctrl ? a.y : b.y;
                }
            }
        }
    } else {
        const int tmask = 1 << bt;
        #pragma unroll
        for (int s = 0; s < 8; ++s) {
            c32 o = shfl_xor_c(st[s], tmask);
            bool ctrl = (bc >= 5) ? (((s >> (bc - 5)) & 1) != 0) : (((lane >> bc) & 1) != 0);
            st[s].x = ctrl ? o.x : st[s].x;
            st[s].y = ctrl ? o.y : st[s].y;
        }
    }
}

// ---------------- K1: half-angle cos/sin table per (circuit, batch) ----------------
__global__ void __launch_bounds__(256)
k_angles(const float* __restrict__ src, float* __restrict__ CS, int layer)
{
    int idx = blockIdx.x * blockDim.x + threadIdx.x;   // 64*256
    int circ = idx >> 8, b = idx & 255;
    int t = circ >> 3, sub = circ & 7;
    float* dst = CS + ((size_t)(circ * 256 + b)) * 16;
    #pragma unroll
    for (int w = 0; w < 8; ++w) {
        float ang;
        if (layer == 0) ang = 0.5f * src[((size_t)b * TSTEPS + t) * 64 + sub * 8 + w]; // x:(B,T,D), block j=sub
        else            ang = 0.5f * src[((size_t)t * NBATCH + b) * 64 + w * 8 + sub]; // H1:(T,B,D), fully mode
        float s, c; sincosf(ang, &s, &c);
        dst[w] = c; dst[8 + w] = s;
    }
}

// ---------------- K2: adjoint row-sim + WMMA GEMM + Z-reduce to partials ----------------
__global__ void __launch_bounds__(256)
k_main(const float* __restrict__ CS, const float* __restrict__ theta,
       float* __restrict__ partial, int layer)
{
    __shared__ float Ur[MT * 257];
    __shared__ float Ui[MT * 257];
    __shared__ float Pt[16 * 256];
    __shared__ float Gt[16 * 8];

    const int circ = blockIdx.x;          // 0..63
    const int kb   = blockIdx.y;          // 0..15
    const int tid  = threadIdx.x;
    const int lane = tid & 31;
    const int wave = tid >> 5;
    const int t_idx = circ >> 3, sub = circ & 7;

    // gate table: transposed Rot entries (T00=U00, T01=U10, T10=U01, T11=U11)
    if (tid < 16) {
        int l = tid >> 3, w = tid & 7;
        size_t base = ((((size_t)t_idx * NLAYERS + layer) * NBLK + sub) * QDEPTH + l) * (NQ * 3) + w * 3;
        float phi = theta[base + 0], th = theta[base + 1], om = theta[base + 2];
        float c = cosf(0.5f * th), s = sinf(0.5f * th);
        float ap = 0.5f * (phi + om), am = 0.5f * (phi - om);
        float cap = cosf(ap), sap = sinf(ap), cam = cosf(am), sam = sinf(am);
        float* g = &Gt[tid * 8];
        g[0] =  cap * c; g[1] = -sap * c;   // T00 = ep*c
        g[2] =  cam * s; g[3] = -sam * s;   // T01 = U10 = em*s
        g[4] = -cam * s; g[5] = -sam * s;   // T10 = U01 = -conj(em)*s
        g[6] =  cap * c; g[7] =  sap * c;   // T11 = conj(ep)*c
    }
    __syncthreads();

    // Phase 1: row k of U = reversed, transposed gate sequence applied to e_k
    for (int rr = 0; rr < 2; ++rr) {
        int m = wave * 2 + rr;
        int krow = kb * MT + m;
        c32 st[8];
        #pragma unroll
        for (int s = 0; s < 8; ++s) { st[s].x = (s * 32 + lane == krow) ? 1.0f : 0.0f; st[s].y = 0.0f; }
        #pragma unroll
        for (int w = 7; w >= 0; --w) cnot_g(st, 7 - w, 7 - ((w + 2) & 7), lane); // l=1 ring (r=2), reversed
        #pragma unroll
        for (int w = 7; w >= 0; --w) rot_g(st, 7 - w, &Gt[(8 + w) * 8], lane);   // l=1 RotT
        #pragma unroll
        for (int w = 7; w >= 0; --w) cnot_g(st, 7 - w, 7 - ((w + 1) & 7), lane); // l=0 ring (r=1), reversed
        #pragma unroll
        for (int w = 7; w >= 0; --w) rot_g(st, 7 - w, &Gt[(0 + w) * 8], lane);   // l=0 RotT
        #pragma unroll
        for (int s = 0; s < 8; ++s) {
            Ur[m * 257 + s * 32 + lane] = st[s].x;
            Ui[m * 257 + s * 32 + lane] = st[s].y;
        }
    }

    // Phase 2: Y(16 x 256) = U_tile(16 x 256) @ P(256 x 256), real & imag, WMMA f32 16x16x4
    v8f accr0 = {}, acci0 = {}, accr1 = {}, acci1 = {};
    const int n0   = wave * 32;            // wave's batch-column range (two 16-wide subtiles)
    const int koff = (lane >> 4) << 1;     // A/B VGPR layout: lanes 16-31 hold K+2,K+3
    const int mA   = lane & 15;
    const float* csg = CS + ((size_t)circ * 256 + tid) * 16;
    float cw[16];
    #pragma unroll
    for (int i = 0; i < 16; ++i) cw[i] = csg[i];

    for (int chunk = 0; chunk < 16; ++chunk) {
        const int k0 = chunk * 16;
        __syncthreads();
        // build P chunk: thread = batch b, 16 consecutive j rows; P[j,b] = prod_w (bit? sin : cos)
        #pragma unroll
        for (int jj = 0; jj < 16; ++jj) {
            int j = k0 + jj;
            float p = 1.0f;
            #pragma unroll
            for (int w = 0; w < 8; ++w) p *= ((j >> (7 - w)) & 1) ? cw[8 + w] : cw[w];
            Pt[jj * 256 + tid] = p;
        }
        __syncthreads();
        #pragma unroll
        for (int ks = 0; ks < 4; ++ks) {
            const int kloc = ks * 4 + koff;
            const int kcol = k0 + kloc;
            v2f ar, ai, b0, b1;
            ar[0] = Ur[mA * 257 + kcol]; ar[1] = Ur[mA * 257 + kcol + 1];
            ai[0] = Ui[mA * 257 + kcol]; ai[1] = Ui[mA * 257 + kcol + 1];
            const int nB0 = n0 + (lane & 15);
            b0[0] = Pt[kloc * 256 + nB0];      b0[1] = Pt[(kloc + 1) * 256 + nB0];
            b1[0] = Pt[kloc * 256 + nB0 + 16]; b1[1] = Pt[(kloc + 1) * 256 + nB0 + 16];
            accr0 = __builtin_amdgcn_wmma_f32_16x16x4_f32(false, ar, false, b0, (short)0, accr0, false, false);
            acci0 = __builtin_amdgcn_wmma_f32_16x16x4_f32(false, ai, false, b0, (short)0, acci0, false, false);
            accr1 = __builtin_amdgcn_wmma_f32_16x16x4_f32(false, ar, false, b1, (short)0, accr1, false, false);
            acci1 = __builtin_amdgcn_wmma_f32_16x16x4_f32(false, ai, false, b1, (short)0, acci1, false, false);
        }
    }

    // probs -> signed Z sums -> deterministic partials
    const int half = lane >> 4;
    #pragma unroll
    for (int sub_t = 0; sub_t < 2; ++sub_t) {
        v8f pr = (sub_t == 0) ? (accr0 * accr0 + acci0 * acci0)
                              : (accr1 * accr1 + acci1 * acci1);
        float ow[8];
        #pragma unroll
        for (int w = 0; w < 8; ++w) ow[w] = 0.0f;
        #pragma unroll
        for (int r = 0; r < 8; ++r) {
            int kglob = kb * MT + r + half * 8;   // C layout: VGPR r -> M=r (+8 for lanes 16-31)
            float p = pr[r];
            #pragma unroll
            for (int w = 0; w < 8; ++w) ow[w] += ((kglob >> (7 - w)) & 1) ? -p : p;
        }
        #pragma unroll
        for (int w = 0; w < 8; ++w) ow[w] += __shfl_xor(ow[w], 16, 32);  // combine the two M-halves
        if (half == 0) {
            int b = n0 + sub_t * 16 + lane;
            float* dst = partial + (((size_t)kb * NCIRC + circ) * NBATCH + b) * 8;
            #pragma unroll
            for (int w = 0; w < 8; ++w) dst[w] = ow[w];
        }
    }
}

// ---------------- K3: fixed-order reduction over k-blocks + layout transpose ----------------
__global__ void __launch_bounds__(256)
k_reduce(const float* __restrict__ partial, float* __restrict__ dst, int layer)
{
    int idx = blockIdx.x * blockDim.x + threadIdx.x;  // 64*256*8
    int circ = idx >> 11;
    int rem = idx & 2047;
    int b = rem >> 3, w = rem & 7;
    float s = 0.0f;
    #pragma unroll
    for (int kb = 0; kb < KBLOCKS; ++kb)
        s += partial[(((size_t)kb * NCIRC + circ) * NBATCH + b) * 8 + w];
    int t = circ >> 3, sub = circ & 7;
    if (layer == 0) dst[((size_t)t * NBATCH + b) * 64 + sub * 8 + w] = s;   // H1: (T,B,D)
    else            dst[((size_t)b * TSTEPS + t) * 64 + sub * 8 + w] = s;   // out: (B,T,D)
}

extern "C" void kernel_launch(void* const* d_in, const int* in_sizes, int n_in,
                              void* d_out, int out_size, void* d_ws, size_t ws_size,
                              hipStream_t stream)
{
    (void)in_sizes; (void)n_in; (void)out_size; (void)ws_size;
    const float* x     = (const float*)d_in[0];   // (B,T,D) f32
    const float* theta = (const float*)d_in[1];   // (T,NL,NB,DEPTH,NQ,3) f32
    float* out = (float*)d_out;                   // (B,T,D) f32

    float* CS      = (float*)d_ws;                // 64*256*16      = 262144 floats
    float* partial = CS + 262144;                 // 16*64*256*8    = 2097152 floats
    float* H1      = partial + 2097152;           // 8*256*64       = 131072 floats (~10 MB total)

    for (int layer = 0; layer < 2; ++layer) {
        const float* src = (layer == 0) ? x : H1;
        float* dst       = (layer == 0) ? H1 : out;
        k_angles<<<64, 256, 0, stream>>>(src, CS, layer);
        k_main<<<dim3(64, 16), 256, 0, stream>>>(CS, theta, partial, layer);
        k_reduce<<<512, 256, 0, stream>>>(partial, dst, layer);
    }
}